// GNN_64750926954676
// MI455X (gfx1250) — compile-verified
//
#include <hip/hip_runtime.h>
#include <math.h>

typedef float v2f __attribute__((ext_vector_type(2)));
typedef float v8f __attribute__((ext_vector_type(8)));

#define NN   100000
#define NE   1600000
#define HID  64

// ---------------- degree counting ----------------
__global__ void count_kernel(const int* __restrict__ dst, int* __restrict__ cnt, int E) {
  int e = blockIdx.x * blockDim.x + threadIdx.x;
  if (e < E) atomicAdd(&cnt[dst[e]], 1);
}

// ---------------- exclusive scan (3 phases) ----------------
__global__ void scan_blocks_kernel(const int* __restrict__ cnt, int* __restrict__ rowptr,
                                   int* __restrict__ blockSums, int total) {
  __shared__ int s[256];
  int tid  = threadIdx.x;
  int base = blockIdx.x * 1024 + tid * 4;
  int v[4]; int sum = 0;
  #pragma unroll
  for (int i = 0; i < 4; ++i) {
    int idx = base + i;
    v[i] = (idx < total) ? cnt[idx] : 0;
    sum += v[i];
  }
  s[tid] = sum;
  __syncthreads();
  for (int off = 1; off < 256; off <<= 1) {
    int t = (tid >= off) ? s[tid - off] : 0;
    __syncthreads();
    s[tid] += t;
    __syncthreads();
  }
  if (tid == 255) blockSums[blockIdx.x] = s[255];
  int run = s[tid] - sum;
  #pragma unroll
  for (int i = 0; i < 4; ++i) {
    int idx = base + i;
    if (idx < total) rowptr[idx] = run;
    run += v[i];
  }
}

__global__ void scan_sums_kernel(int* __restrict__ blockSums, int nblocks) {
  if (threadIdx.x == 0 && blockIdx.x == 0) {
    int run = 0;
    for (int b = 0; b < nblocks; ++b) {
      int t = blockSums[b];
      blockSums[b] = run;
      run += t;
    }
  }
}

__global__ void add_offsets_kernel(int* __restrict__ rowptr, const int* __restrict__ blockSums,
                                   int total) {
  int idx = blockIdx.x * blockDim.x + threadIdx.x;
  if (idx < total) rowptr[idx] += blockSums[idx >> 10];
}

// ---------------- dinv = rsqrt(deg + 1)  (self-loop) ----------------
__global__ void dinv_kernel(const int* __restrict__ cnt, float* __restrict__ dinv, int n) {
  int i = blockIdx.x * blockDim.x + threadIdx.x;
  if (i < n) dinv[i] = rsqrtf((float)cnt[i] + 1.0f);
}

// ---------------- CSR fill ----------------
__global__ void fill_kernel(const int* __restrict__ src, const int* __restrict__ dst,
                            const int* __restrict__ rowptr, int* __restrict__ cursor,
                            int* __restrict__ esrc, int E) {
  int e = blockIdx.x * blockDim.x + threadIdx.x;
  if (e < E) {
    int d   = dst[e];
    int off = atomicAdd(&cursor[d], 1);
    esrc[rowptr[d] + off] = src[e];
  }
}

// ---------------- per-row sort: makes neighbor order (and f32 sums) deterministic --------
__global__ void sort_rows_kernel(int* __restrict__ esrc, const int* __restrict__ rowptr, int n) {
  int i = blockIdx.x * blockDim.x + threadIdx.x;
  if (i >= n) return;
  int beg = rowptr[i], end = rowptr[i + 1];
  for (int a = beg + 1; a < end; ++a) {
    int key = esrc[a];
    int b   = a - 1;
    while (b >= beg && esrc[b] > key) { esrc[b + 1] = esrc[b]; --b; }
    esrc[b + 1] = key;
  }
}

// ---------------- per-edge weight ew[j] = dinv[src_j] (dst factor hoisted out) ----------
__global__ void ew_kernel(const int* __restrict__ esrc, const float* __restrict__ dinv,
                          float* __restrict__ ew, int E) {
  int j = blockIdx.x * blockDim.x + threadIdx.x;
  if (j < E) ew[j] = dinv[esrc[j]];
}

// ---------------- WMMA GEMM:  out[M x 64] = A[M x K] @ W[64 x K]^T + bias ----------------
// One wave32 computes a 16-row x 64-col output stripe using V_WMMA_F32_16X16X4_F32.
// W is staged in LDS with a +2 float row pad so B-fragment reads are bank-conflict free.
__global__ __launch_bounds__(256)
void gemm_wmma_kernel(const float* __restrict__ A, const float* __restrict__ W,
                      const float* __restrict__ bias, float* __restrict__ out,
                      int M, int K) {
  __shared__ float sW[64 * 130];           // supports K up to 128 with pad of 2
  const int stride = K + 2;
  int tid = threadIdx.x;
  for (int i = tid; i < 64 * K; i += 256) {
    int row = i / K, col = i - row * K;
    sW[row * stride + col] = W[i];
  }
  __syncthreads();

  int wave = tid >> 5;
  int lane = tid & 31;
  int r    = lane & 15;      // row-in-tile for A, col-in-tile for B/C
  int kh   = lane >> 4;      // which K-half of the fragment

  int tile  = blockIdx.x * 8 + wave;       // wave-uniform guard keeps EXEC all-1s
  int tiles = M >> 4;
  if (tile >= tiles) return;

  int m = tile * 16 + r;
  const float* arow = A + (size_t)m * K;

  v8f c0 = {}, c1 = {}, c2 = {}, c3 = {};
  for (int k0 = 0; k0 < K; k0 += 4) {
    int kc = k0 + 2 * kh;
    v2f a  = *(const v2f*)(arow + kc);                   // A frag: 16x4 f32
    v2f b0 = *(const v2f*)(&sW[(r +  0) * stride + kc]); // B frag: 4x16 (B[k][n]=W[n][k])
    v2f b1 = *(const v2f*)(&sW[(r + 16) * stride + kc]);
    v2f b2 = *(const v2f*)(&sW[(r + 32) * stride + kc]);
    v2f b3 = *(const v2f*)(&sW[(r + 48) * stride + kc]);
    c0 = __builtin_amdgcn_wmma_f32_16x16x4_f32(false, a, false, b0, (short)0, c0, false, false);
    c1 = __builtin_amdgcn_wmma_f32_16x16x4_f32(false, a, false, b1, (short)0, c1, false, false);
    c2 = __builtin_amdgcn_wmma_f32_16x16x4_f32(false, a, false, b2, (short)0, c2, false, false);
    c3 = __builtin_amdgcn_wmma_f32_16x16x4_f32(false, a, false, b3, (short)0, c3, false, false);
  }

  // C/D layout: VGPR j -> M = j + 8*kh, N = lane&15
  int row0 = tile * 16;
  float bb0 = bias[r], bb1 = bias[r + 16], bb2 = bias[r + 32], bb3 = bias[r + 48];
  #pragma unroll
  for (int j = 0; j < 8; ++j) {
    float* orow = out + (size_t)(row0 + j + 8 * kh) * 64;
    orow[r]      = c0[j] + bb0;
    orow[r + 16] = c1[j] + bb1;
    orow[r + 32] = c2[j] + bb2;
    orow[r + 48] = c3[j] + bb3;
  }
}

// ---------------- propagation + L2-normalize + ReLU (one wave32 per node) ----------------
// agg = di * sum_j ew[j]*h[src_j]  +  di^2 * h[node];  out = 0.5*h + 0.5*agg (APPNP consts)
// Each lane owns features {2*lane, 2*lane+1} -> float2 (global_load_b64) per gather.
__global__ __launch_bounds__(256)
void prop_kernel(const float* __restrict__ hin, float* __restrict__ hout,
                 const int* __restrict__ rowptr, const int* __restrict__ esrc,
                 const float* __restrict__ ew, const float* __restrict__ dinv, int n) {
  int wave = threadIdx.x >> 5;
  int lane = threadIdx.x & 31;
  int node = blockIdx.x * 8 + wave;
  if (node >= n) return;

  const float2* hin2 = (const float2*)hin;
  float2*       hout2 = (float2*)hout;

  int beg = rowptr[node], end = rowptr[node + 1];

  float a0 = 0.f, a1 = 0.f;
  for (int j = beg; j < end; ++j) {
    int    s  = esrc[j];
    float  w  = ew[j];
    float2 hv = hin2[(size_t)s * 32 + lane];
    a0 = fmaf(w, hv.x, a0);
    a1 = fmaf(w, hv.y, a1);
  }
  float  di = dinv[node];
  float  d2 = di * di;
  float2 hm = hin2[(size_t)node * 32 + lane];
  a0 = fmaf(d2, hm.x, a0 * di);          // di * (edge sum) + di^2 * self
  a1 = fmaf(d2, hm.y, a1 * di);

  const float ALPHA = 0.5f, ALPHA_RES = 0.0f, BETA = 1.0f;
  float o0 = ALPHA_RES * hm.x + (1.f - ALPHA_RES) * (ALPHA * hm.x + (1.f - ALPHA) * BETA * a0);
  float o1 = ALPHA_RES * hm.y + (1.f - ALPHA_RES) * (ALPHA * hm.y + (1.f - ALPHA) * BETA * a1);

  // L2 normalize across the 64 features (wave32 reduction) then ReLU
  float ss = o0 * o0 + o1 * o1;
  #pragma unroll
  for (int off = 16; off > 0; off >>= 1) ss += __shfl_xor(ss, off, 32);
  float inv = 1.f / fmaxf(sqrtf(ss), 1e-12f);
  o0 = fmaxf(o0 * inv, 0.f);
  o1 = fmaxf(o1 * inv, 0.f);

  float2 o; o.x = o0; o.y = o1;
  hout2[(size_t)node * 32 + lane] = o;
}

// ---------------- launch ----------------
extern "C" void kernel_launch(void* const* d_in, const int* in_sizes, int n_in,
                              void* d_out, int out_size, void* d_ws, size_t ws_size,
                              hipStream_t stream) {
  const float* x  = (const float*)d_in[0];
  const int*   ei = (const int*)d_in[1];
  const float* W0 = (const float*)d_in[2];
  const float* b0 = (const float*)d_in[3];
  const float* W1 = (const float*)d_in[4];
  const float* b1 = (const float*)d_in[5];
  const float* W2 = (const float*)d_in[6];
  const float* b2 = (const float*)d_in[7];
  const int* src = ei;           // edge_index[0]
  const int* dst = ei + NE;      // edge_index[1]
  float* out = (float*)d_out;

  // workspace carve (aligned 256B)
  char*  ws  = (char*)d_ws;
  size_t off = 0;
  auto carve = [&](size_t bytes) -> void* {
    void* p = ws + off;
    off = (off + bytes + 255) & ~(size_t)255;
    return p;
  };
  float* hB        = (float*)carve(sizeof(float) * (size_t)NN * 64);
  int*   esrc      = (int*)  carve(sizeof(int) * (size_t)NE);
  float* ew        = (float*)carve(sizeof(float) * (size_t)NE);
  int*   cnt       = (int*)  carve(sizeof(int) * (NN + 1));
  int*   rowptr    = (int*)  carve(sizeof(int) * (NN + 1));
  int*   cursor    = (int*)  carve(sizeof(int) * NN);
  float* dinv      = (float*)carve(sizeof(float) * NN);
  int*   blockSums = (int*)  carve(sizeof(int) * 128);

  hipMemsetAsync(cnt,    0, sizeof(int) * (NN + 1), stream);
  hipMemsetAsync(cursor, 0, sizeof(int) * NN,       stream);

  // ----- build CSR + weights (shared by both propagation layers) -----
  int ethreads = 256, eblocks = (NE + 255) / 256;
  count_kernel<<<eblocks, ethreads, 0, stream>>>(dst, cnt, NE);

  int total   = NN + 1;
  int sblocks = (total + 1023) / 1024;     // 98
  scan_blocks_kernel<<<sblocks, 256, 0, stream>>>(cnt, rowptr, blockSums, total);
  scan_sums_kernel<<<1, 32, 0, stream>>>(blockSums, sblocks);
  add_offsets_kernel<<<(total + 255) / 256, 256, 0, stream>>>(rowptr, blockSums, total);

  int nblocks = (NN + 255) / 256;
  dinv_kernel<<<nblocks, 256, 0, stream>>>(cnt, dinv, NN);
  fill_kernel<<<eblocks, ethreads, 0, stream>>>(src, dst, rowptr, cursor, esrc, NE);
  sort_rows_kernel<<<nblocks, 256, 0, stream>>>(esrc, rowptr, NN);
  ew_kernel<<<eblocks, ethreads, 0, stream>>>(esrc, dinv, ew, NE);

  // ----- layer 0: GEMM (WMMA f32) -> prop+norm+relu -----
  int gtiles  = NN / 16;                   // 6250 (NN divisible by 16)
  int gblocks = (gtiles + 7) / 8;          // 8 waves per block
  int pblocks = (NN + 7) / 8;              // one wave per node

  gemm_wmma_kernel<<<gblocks, 256, 0, stream>>>(x, W0, b0, out, NN, 128);
  prop_kernel<<<pblocks, 256, 0, stream>>>(out, hB, rowptr, esrc, ew, dinv, NN);

  // ----- layer 1 -----
  gemm_wmma_kernel<<<gblocks, 256, 0, stream>>>(hB, W1, b1, out, NN, 64);
  prop_kernel<<<pblocks, 256, 0, stream>>>(out, hB, rowptr, esrc, ew, dinv, NN);

  // ----- final linear -----
  gemm_wmma_kernel<<<gblocks, 256, 0, stream>>>(hB, W2, b2, out, NN, 64);
}